// axon_layer_14946486190129
// MI455X (gfx1250) — compile-verified
//
#include <hip/hip_runtime.h>

typedef __attribute__((ext_vector_type(2))) float v2f;
typedef __attribute__((ext_vector_type(4))) float v4f;
typedef __attribute__((ext_vector_type(8))) float v8f;

#define B_   32
#define F_   4096
#define T_   400
#define TCH  16           // timesteps per WMMA chunk
#define WAVES_PER_BLOCK 8

// exact integer power by repeated multiply (setup-only, <=16 iters)
__device__ __forceinline__ float ipow(float d, int p) {
    float r = 1.0f;
    for (int k = 0; k < p; ++k) r *= d;
    return r;
}

__global__ __launch_bounds__(256) void axon_psp_wmma(
    const float* __restrict__ xin_g,   // input_spikes [B,F,T]
    const float* __restrict__ dm_v,    // decay_m [F]
    const float* __restrict__ ds_v,    // decay_s [F]
    const float* __restrict__ v0_p,    // v_0 scalar
    float* __restrict__ out)           // psps [B,F,T] ++ m_T [B,F] ++ s_T [B,F]
{
    const int lane = threadIdx.x & 31;
    const int wv   = blockIdx.x * WAVES_PER_BLOCK + (threadIdx.x >> 5);
    const int f    = wv >> 1;          // feature for this wave
    const int b0   = (wv & 1) << 4;    // batch tile base: 0 or 16

    const int n = lane & 15;           // column (batch) for B/C/D; row j for A
    const int h = lane >> 4;           // lane half

    const float dm = dm_v[f];
    const float ds = ds_v[f];
    const float v0 = v0_p[0];

    // ---- A operand: G[j,i] = v0*(dm^(j-i) - ds^(j-i)) for i<j, else 0.
    // A-layout (16x4 f32): lane -> M = lane%16, element e -> K = 2h+e; wmma c adds 4c.
    v2f A[4];
#pragma unroll
    for (int c = 0; c < 4; ++c) {
#pragma unroll
        for (int e = 0; e < 2; ++e) {
            const int i = 4 * c + 2 * h + e;
            const int p = n - i;                         // j - i, j == n for A
            A[c][e] = (p > 0) ? v0 * (ipow(dm, p) - ipow(ds, p)) : 0.0f;
        }
    }

    // ---- carry-bias coefficients for C init: rows j = 8h + v
    float Bm[8], Bs[8];
#pragma unroll
    for (int v = 0; v < 8; ++v) {
        const int j = 8 * h + v;
        Bm[v] = v0 * ipow(dm, j + 1);
        Bs[v] = v0 * ipow(ds, j + 1);
    }

    // ---- state-closure coefficients: dm^(15-i) for this lane's x elements
    float Sm[4][2], Ss[4][2];
#pragma unroll
    for (int c = 0; c < 4; ++c)
#pragma unroll
        for (int e = 0; e < 2; ++e) {
            const int i = 4 * c + 2 * h + e;
            Sm[c][e] = ipow(dm, 15 - i);
            Ss[c][e] = ipow(ds, 15 - i);
        }
    const float dm16 = ipow(dm, 16);
    const float ds16 = ipow(ds, 16);

    // this lane's batch row
    const long row = ((long)(b0 + n) * F_ + f) * T_;
    const float* xin  = xin_g + row;
    float*       pout = out   + row;

    float m = 0.0f, s = 0.0f;

    for (int t0 = 0; t0 < T_; t0 += TCH) {
        // ---- load B operand: lane holds X[i = 4c+2h+e, n] as v2f per wmma slice
        v2f xv[4];
#pragma unroll
        for (int c = 0; c < 4; ++c)
            xv[c] = *(const v2f*)(xin + t0 + 4 * c + 2 * h);

        // prefetch next chunk of this row (global_prefetch_b8, speculative)
        __builtin_prefetch(xin + t0 + TCH + 2 * h, 0, 1);

        // ---- init accumulator with rank-1 carry bias
        v8f acc;
#pragma unroll
        for (int v = 0; v < 8; ++v) acc[v] = Bm[v] * m - Bs[v] * s;

        // ---- 4 chained f32 WMMAs: acc += G * X   (16x16x4 each)
#pragma unroll
        for (int c = 0; c < 4; ++c) {
            acc = __builtin_amdgcn_wmma_f32_16x16x4_f32(
                /*neg_a=*/false, A[c], /*neg_b=*/false, xv[c],
                /*c_mod=*/(short)0, acc, /*reuse_a=*/false, /*reuse_b=*/false);
        }

        // ---- store psp: lane's rows j = 8h..8h+7 are 8 consecutive t (32B)
        v4f lo = { acc[0], acc[1], acc[2], acc[3] };
        v4f hi = { acc[4], acc[5], acc[6], acc[7] };
        *(v4f*)(pout + t0 + 8 * h)     = lo;
        *(v4f*)(pout + t0 + 8 * h + 4) = hi;

        // ---- state closure: per-column dot + cross-half reduce
        float pm = 0.0f, ps = 0.0f;
#pragma unroll
        for (int c = 0; c < 4; ++c) {
            pm += Sm[c][0] * xv[c][0] + Sm[c][1] * xv[c][1];
            ps += Ss[c][0] * xv[c][0] + Ss[c][1] * xv[c][1];
        }
        pm += __shfl_xor(pm, 16, 32);
        ps += __shfl_xor(ps, 16, 32);
        m = dm16 * m + pm;
        s = ds16 * s + ps;
    }

    // ---- final states (both halves hold identical state; half 0 writes)
    if (h == 0) {
        const long base = (long)B_ * F_ * T_;
        out[base + (long)(b0 + n) * F_ + f]                    = m;
        out[base + (long)B_ * F_ + (long)(b0 + n) * F_ + f]    = s;
    }
}

extern "C" void kernel_launch(void* const* d_in, const int* in_sizes, int n_in,
                              void* d_out, int out_size, void* d_ws, size_t ws_size,
                              hipStream_t stream) {
    const float* x   = (const float*)d_in[0];   // input_spikes [32,4096,400]
    const float* dm  = (const float*)d_in[1];   // decay_m [4096]
    const float* ds  = (const float*)d_in[2];   // decay_s [4096]
    const float* v0  = (const float*)d_in[3];   // v_0 scalar
    float*       out = (float*)d_out;

    // waves = F * (B/16) = 8192; 8 waves/block -> 1024 blocks of 256 threads
    const int blocks = (F_ * (B_ / 16)) / WAVES_PER_BLOCK;
    axon_psp_wmma<<<blocks, WAVES_PER_BLOCK * 32, 0, stream>>>(x, dm, ds, v0, out);
}